// BiMambaMSM_5050881540707
// MI455X (gfx1250) — compile-verified
//
#include <hip/hip_runtime.h>
#include <hip/hip_bf16.h>
#include <math.h>

typedef __attribute__((ext_vector_type(2))) float v2f;
typedef __attribute__((ext_vector_type(4))) float v4f;
typedef __attribute__((ext_vector_type(8))) float v8f;

#define B_      2
#define L_      2048
#define DIN     80
#define DM      512
#define DI_     1024
#define DS_     64
#define NH_     16
#define CONVC_  1152
#define DPROJ_  2192
#define ROWS    (B_ * L_)   // 4096

// ---------------------------------------------------------------------------
// GEMM: C[M,N] = A[M,K] @ B[K,N] + bias[N]   (f32 WMMA 16x16x4)
// One wave computes a 64x16 strip: 4 M-tiles sharing one B fragment
// (6 VMEM per 4 WMMAs, 4 independent accumulation chains).
// Block = 8 waves, 2(M-groups of 64) x 4(N) -> 128x64 block tile.
// A fragment (16x4): lane<16 holds row m, {K0,K1}; lane>=16 holds {K2,K3}.
// B fragment (4x16): lane<16 holds col n, {K0,K1}; lane>=16 holds {K2,K3}.
// D: VGPR r -> row (r + 8*(lane/16)), col lane%16.
// ---------------------------------------------------------------------------
__global__ __launch_bounds__(256) void gemm_wmma_f32(
    const float* __restrict__ A, const float* __restrict__ Bm,
    const float* __restrict__ bias, float* __restrict__ C,
    int M, int N, int K, int lda, int ldb, int ldc) {
  const int tid  = threadIdx.x;
  const int wave = tid >> 5;
  const int lane = tid & 31;
  const int wm   = wave >> 2;                       // 0..1
  const int wn   = wave & 3;                        // 0..3
  const int row0 = blockIdx.y * 128 + wm * 64;      // wave's 64-row strip
  const int col0 = (blockIdx.x * 4 + wn) * 16;
  if (row0 >= M || col0 >= N) return;

  const int half = lane >> 4;                       // K-half select
  const int l16  = lane & 15;
  const int n  = col0 + l16;                        // B col held by this lane
  const int kh = 2 * half;

  v8f c0, c1, c2, c3;
  const float bv = bias ? bias[n] : 0.0f;
#pragma unroll
  for (int r = 0; r < 8; ++r) { c0[r] = bv; c1[r] = bv; c2[r] = bv; c3[r] = bv; }

  const float* a0 = A + (size_t)(row0 +  0 + l16) * lda;
  const float* a1 = A + (size_t)(row0 + 16 + l16) * lda;
  const float* a2 = A + (size_t)(row0 + 32 + l16) * lda;
  const float* a3 = A + (size_t)(row0 + 48 + l16) * lda;

  for (int k0 = 0; k0 < K; k0 += 4) {
    const int ka = k0 + kh;
    v2f b;
    b.x = Bm[(size_t)ka * ldb + n];
    b.y = Bm[(size_t)(ka + 1) * ldb + n];
    v2f f0 = *(const v2f*)(a0 + ka);
    v2f f1 = *(const v2f*)(a1 + ka);
    v2f f2 = *(const v2f*)(a2 + ka);
    v2f f3 = *(const v2f*)(a3 + ka);
    c0 = __builtin_amdgcn_wmma_f32_16x16x4_f32(false, f0, false, b, (short)0, c0, false, false);
    c1 = __builtin_amdgcn_wmma_f32_16x16x4_f32(false, f1, false, b, (short)0, c1, false, false);
    c2 = __builtin_amdgcn_wmma_f32_16x16x4_f32(false, f2, false, b, (short)0, c2, false, false);
    c3 = __builtin_amdgcn_wmma_f32_16x16x4_f32(false, f3, false, b, (short)0, c3, false, false);
  }

#pragma unroll
  for (int r = 0; r < 8; ++r) {
    const int rr = r + 8 * half;
    C[(size_t)(row0 +  0 + rr) * ldc + n] = c0[r];
    C[(size_t)(row0 + 16 + rr) * ldc + n] = c1[r];
    C[(size_t)(row0 + 32 + rr) * ldc + n] = c2[r];
    C[(size_t)(row0 + 48 + rr) * ldc + n] = c3[r];
  }
}

// ---------------------------------------------------------------------------
// Depthwise causal conv (fwd) and anticausal conv (bwd-aligned) + SiLU.
// xBC lives at zxbcdt columns [DI, DI+CONVC).
// fwd:  y[l] = b + sum_k w[k]*x[l-6+k];  bwd-aligned: y[l] = b + sum_k w[k]*x[l+6-k]
// ---------------------------------------------------------------------------
__global__ __launch_bounds__(256) void conv_silu_kernel(
    const float* __restrict__ zx, const float* __restrict__ conv_w,
    const float* __restrict__ conv_b, float* __restrict__ xcf,
    float* __restrict__ xcb) {
  int idx = blockIdx.x * 256 + threadIdx.x;
  if (idx >= ROWS * CONVC_) return;
  int c   = idx % CONVC_;
  int row = idx / CONVC_;
  int l   = row % L_;
  int b   = row / L_;
  const float* src = zx + (size_t)(b * L_) * DPROJ_ + DI_ + c;
  float w[7];
#pragma unroll
  for (int k = 0; k < 7; ++k) w[k] = conv_w[c * 7 + k];
  float accf = conv_b[c], accb = conv_b[c];
#pragma unroll
  for (int k = 0; k < 7; ++k) {
    int tf = l - 6 + k;
    if (tf >= 0) accf += w[k] * src[(size_t)tf * DPROJ_];
    int tb = l + 6 - k;
    if (tb < L_) accb += w[k] * src[(size_t)tb * DPROJ_];
  }
  xcf[idx] = accf / (1.0f + __expf(-accf));
  xcb[idx] = accb / (1.0f + __expf(-accb));
}

// dt = softplus(raw + dt_bias);  dA = exp(-exp(A_log) * dt)   (direction-shared)
__global__ __launch_bounds__(256) void dt_kernel(
    const float* __restrict__ zx, const float* __restrict__ dt_bias,
    const float* __restrict__ A_log, float* __restrict__ dt_out,
    float* __restrict__ dA_out) {
  int idx = blockIdx.x * 256 + threadIdx.x;
  if (idx >= ROWS * NH_) return;
  int h   = idx % NH_;
  int row = idx / NH_;
  float x  = zx[(size_t)row * DPROJ_ + DI_ + CONVC_ + h] + dt_bias[h];
  float dt = (x > 20.0f) ? x : log1pf(__expf(x));
  dt_out[idx] = dt;
  dA_out[idx] = __expf(-__expf(A_log[h]) * dt);
}

// ---------------------------------------------------------------------------
// Selective scan, both directions in one launch (64 blocks: dir = blk>>5).
// One block per (dir, batch, head). Thread owns p = tid/4, n = (tid&3)*16..+15.
// No LDS, no barriers: per-step B/C slices come in as b128 loads (8 lanes per
// wave share each address -> L0 broadcast). Register double-buffering hides
// the load latency of step t+1 behind the FMA chain of step t. The only
// cross-lane op is the 4-lane shfl_xor reduction for the C contraction.
// ---------------------------------------------------------------------------
__global__ __launch_bounds__(256) void scan_kernel(
    const float* __restrict__ xcf, const float* __restrict__ xcb,
    const float* __restrict__ dt, const float* __restrict__ dA,
    const float* __restrict__ Dpar, float* __restrict__ yfp,
    float* __restrict__ ybp) {
  const int blk = blockIdx.x;            // 0..63
  const int dir = blk >> 5;              // 0 fwd, 1 bwd
  const int bh  = blk & 31;
  const int b   = bh >> 4;
  const int h   = bh & 15;
  const float* __restrict__ xconv = dir ? xcb : xcf;
  float* __restrict__ yraw        = dir ? ybp : yfp;

  const int tid = threadIdx.x;
  const int p   = tid >> 2;
  const int g   = tid & 3;
  const int n0  = g * 16;

  float s[16];
#pragma unroll
  for (int j = 0; j < 16; ++j) s[j] = 0.0f;
  const float dp = Dpar[h];

  // current-step staged operands
  v4f Bc[4], Cc[4];
  float xc, dAc, dtc;
  size_t row_cur = (size_t)b * L_ + (dir ? (L_ - 1) : 0);
  {
    const float* xr = xconv + row_cur * CONVC_;
#pragma unroll
    for (int q = 0; q < 4; ++q) {
      Bc[q] = *(const v4f*)(xr + DI_ + n0 + q * 4);
      Cc[q] = *(const v4f*)(xr + DI_ + DS_ + n0 + q * 4);
    }
    xc  = xr[h * 64 + p];
    dAc = dA[row_cur * NH_ + h];
    dtc = dt[row_cur * NH_ + h];
  }

  for (int t = 0; t < L_; ++t) {
    // issue loads for step t+1 first (latency hidden behind compute below)
    v4f Bn[4], Cn[4];
    float xn = 0.0f, dAn = 0.0f, dtn = 0.0f;
    size_t row_nxt = row_cur;
    const bool have = (t + 1 < L_);
    if (have) {
      const int ln = dir ? (L_ - 2 - t) : (t + 1);
      row_nxt = (size_t)b * L_ + ln;
      const float* xrn = xconv + row_nxt * CONVC_;
#pragma unroll
      for (int q = 0; q < 4; ++q) {
        Bn[q] = *(const v4f*)(xrn + DI_ + n0 + q * 4);
        Cn[q] = *(const v4f*)(xrn + DI_ + DS_ + n0 + q * 4);
      }
      xn  = xrn[h * 64 + p];
      dAn = dA[row_nxt * NH_ + h];
      dtn = dt[row_nxt * NH_ + h];
    }

    // state update + C contraction for step t
    const float coef = dtc * xc;
    float part = 0.0f;
#pragma unroll
    for (int q = 0; q < 4; ++q) {
#pragma unroll
      for (int e = 0; e < 4; ++e) {
        const int j = q * 4 + e;
        s[j] = fmaf(dAc, s[j], coef * Bc[q][e]);
        part = fmaf(s[j], Cc[q][e], part);
      }
    }
    part += __shfl_xor(part, 1);
    part += __shfl_xor(part, 2);
    if (g == 0) yraw[row_cur * DI_ + h * 64 + p] = part + dp * xc;

    if (have) {
#pragma unroll
      for (int q = 0; q < 4; ++q) { Bc[q] = Bn[q]; Cc[q] = Cn[q]; }
      xc = xn; dAc = dAn; dtc = dtn;
      row_cur = row_nxt;
    }
  }
}

// ---------------------------------------------------------------------------
// Fused gate (SiLU(z)) + per-direction RMSNorm + direction sum.
// Writes n_sum into the dead xBC region of zxbcdt (ld = DPROJ_).
// ---------------------------------------------------------------------------
__global__ __launch_bounds__(256) void norm_kernel(
    const float* __restrict__ zx, const float* __restrict__ yf,
    const float* __restrict__ yb, const float* __restrict__ norm_w,
    float* __restrict__ nsum) {
  const int row = blockIdx.x;
  const int tid = threadIdx.x;
  const float* zr = zx + (size_t)row * DPROJ_;
  float gf[4], gb[4];
  float ssf = 0.0f, ssb = 0.0f;
#pragma unroll
  for (int i = 0; i < 4; ++i) {
    int c = tid + i * 256;
    float z  = zr[c];
    float sz = z / (1.0f + __expf(-z));
    float a  = yf[(size_t)row * DI_ + c] * sz;
    float bb = yb[(size_t)row * DI_ + c] * sz;
    gf[i] = a; gb[i] = bb;
    ssf += a * a; ssb += bb * bb;
  }
  for (int off = 16; off > 0; off >>= 1) {
    ssf += __shfl_down(ssf, off);
    ssb += __shfl_down(ssb, off);
  }
  __shared__ float rf[8], rb[8];
  __shared__ float msF, msB;
  const int wave = tid >> 5, lane = tid & 31;
  if (lane == 0) { rf[wave] = ssf; rb[wave] = ssb; }
  __syncthreads();
  if (tid == 0) {
    float a = 0.0f, b2 = 0.0f;
    for (int w = 0; w < 8; ++w) { a += rf[w]; b2 += rb[w]; }
    msF = a; msB = b2;
  }
  __syncthreads();
  const float rsf = rsqrtf(msF * (1.0f / 1024.0f) + 1e-5f);
  const float rsb = rsqrtf(msB * (1.0f / 1024.0f) + 1e-5f);
#pragma unroll
  for (int i = 0; i < 4; ++i) {
    int c = tid + i * 256;
    nsum[(size_t)row * DPROJ_ + c] = (gf[i] * rsf + gb[i] * rsb) * norm_w[c];
  }
}

// ---------------------------------------------------------------------------
extern "C" void kernel_launch(void* const* d_in, const int* in_sizes, int n_in,
                              void* d_out, int out_size, void* d_ws, size_t ws_size,
                              hipStream_t stream) {
  const float* x       = (const float*)d_in[0];
  const float* Wp_in   = (const float*)d_in[1];
  const float* bp_in   = (const float*)d_in[2];
  const float* W_in    = (const float*)d_in[3];
  const float* b_in    = (const float*)d_in[4];
  const float* conv_w  = (const float*)d_in[5];
  const float* conv_b  = (const float*)d_in[6];
  const float* dt_bias = (const float*)d_in[7];
  const float* A_log   = (const float*)d_in[8];
  const float* Dpar    = (const float*)d_in[9];
  const float* norm_w  = (const float*)d_in[10];
  const float* W_out   = (const float*)d_in[11];
  const float* Wp_out  = (const float*)d_in[12];
  const float* bp_out  = (const float*)d_in[13];
  float* out = (float*)d_out;

  float* ws = (float*)d_ws;
  size_t off = 0;
  float* h     = ws + off; off += (size_t)ROWS * DM;      // reused for t later
  float* zx    = ws + off; off += (size_t)ROWS * DPROJ_;
  float* xcf   = ws + off; off += (size_t)ROWS * CONVC_;
  float* xcb   = ws + off; off += (size_t)ROWS * CONVC_;
  float* dtbuf = ws + off; off += (size_t)ROWS * NH_;
  float* dAbuf = ws + off; off += (size_t)ROWS * NH_;
  float* yf    = ws + off; off += (size_t)ROWS * DI_;
  float* yb    = ws + off; off += (size_t)ROWS * DI_;

  dim3 blk(256);

  // 1) h = x @ Wp_in + bp_in                       (4096x512, K=80)
  gemm_wmma_f32<<<dim3((DM + 63) / 64, ROWS / 128), blk, 0, stream>>>(
      x, Wp_in, bp_in, h, ROWS, DM, DIN, DIN, DM, DM);

  // 2) zxbcdt = h @ W_in + b_in                    (4096x2192, K=512)
  gemm_wmma_f32<<<dim3((DPROJ_ + 63) / 64, ROWS / 128), blk, 0, stream>>>(
      h, W_in, b_in, zx, ROWS, DPROJ_, DM, DM, DPROJ_, DPROJ_);

  // 3) dual-direction depthwise conv + SiLU
  conv_silu_kernel<<<dim3((ROWS * CONVC_) / 256), blk, 0, stream>>>(
      zx, conv_w, conv_b, xcf, xcb);

  // 4) dt / dA precompute (shared by both directions)
  dt_kernel<<<dim3((ROWS * NH_) / 256), blk, 0, stream>>>(
      zx, dt_bias, A_log, dtbuf, dAbuf);

  // 5) selective scans, both directions concurrently (64 blocks)
  scan_kernel<<<dim3(2 * B_ * NH_), blk, 0, stream>>>(
      xcf, xcb, dtbuf, dAbuf, Dpar, yf, yb);

  // 6) gate + RMSNorm + direction sum -> zx cols [DI, 2*DI)
  norm_kernel<<<dim3(ROWS), blk, 0, stream>>>(zx, yf, yb, norm_w, zx + DI_);

  // 7) t = n_sum @ W_out                           (4096x512, K=1024) -> reuse h
  gemm_wmma_f32<<<dim3((DM + 63) / 64, ROWS / 128), blk, 0, stream>>>(
      zx + DI_, W_out, nullptr, h, ROWS, DM, DI_, DPROJ_, DM, DM);

  // 8) out = t @ Wp_out + bp_out                   (4096x80, K=512)
  gemm_wmma_f32<<<dim3((DIN + 63) / 64, ROWS / 128), blk, 0, stream>>>(
      h, Wp_out, bp_out, out, ROWS, DIN, DM, DM, DIN, DIN);
}